// StrongEntanglerLayer_2448131359096
// MI455X (gfx1250) — compile-verified
//
#include <hip/hip_runtime.h>

typedef float v2f __attribute__((ext_vector_type(2)));
typedef float v8f __attribute__((ext_vector_type(8)));
typedef unsigned int u32x4 __attribute__((ext_vector_type(4)));
typedef int i32x8 __attribute__((ext_vector_type(8)));
typedef int i32x4 __attribute__((ext_vector_type(4)));

#define NQ 20
#define BATCH 32
#define MASK20 0xFFFFFu

// perm(y) = y ^ (y>>1) ^ ((y&1) ? bits 19,18 : 0)   (CNOT-ring, qubit0 = MSB)
__device__ __forceinline__ unsigned permL(unsigned y) {
  return (y ^ (y >> 1)) ^ ((y & 1u) ? 0xC0000u : 0u);
}
// f^T(u): bit0 ^= bit19, then prefix-XOR from LSB, mask to 20 bits
__device__ __forceinline__ unsigned fTmap(unsigned u) {
  unsigned w = u ^ ((u >> 19) & 1u);
  unsigned r = w;
  r ^= r << 1; r ^= r << 2; r ^= r << 4; r ^= r << 8; r ^= r << 16;
  return r & MASK20;
}

struct c32 { float x, y; };
__device__ __forceinline__ c32 cmul(c32 a, c32 b) {
  return { a.x * b.x - a.y * b.y, a.x * b.y + a.y * b.x };
}

// ---------------------------------------------------------------------------
// Tensor Data Mover: one D# descriptor moving a (dim0 x dim1) tile of 8-byte
// elements between global and LDS (tile == tensor, so never OOB).
// D# layout per CDNA5 ISA ch.8: group0 {count|lds_addr|global_addr|type=2},
// group1 {data_size=3, dims/strides}, groups 2/3 benign (unused dims = 1/0).
// This toolchain uses the 6-operand builtin form; the trailing i32x8 is
// zero-filled. Issue from one wave; completion via TENSORcnt.
// ---------------------------------------------------------------------------
__device__ __forceinline__ void tdm_move(bool store, unsigned ldsOff,
                                         const void* g, unsigned dim0,
                                         unsigned dim1, unsigned stride0) {
  unsigned long long ga = (unsigned long long)(size_t)g;
  u32x4 g0 = { 1u,                                  // count=1, user mode
               ldsOff,                              // LDS byte address
               (unsigned)ga,                        // global_addr[31:0]
               (unsigned)((ga >> 32) & 0x1FFFFFFu) | (2u << 30) }; // +type=2
  i32x8 g1;
  g1[0] = 0x00030000;                               // data_size = 8B
  g1[1] = (int)((dim0 & 0xFFFFu) << 16);            // tensor_dim0[15:0]
  g1[2] = (int)(((dim0 >> 16) & 0xFFFFu) | ((dim1 & 0xFFFFu) << 16));
  g1[3] = (int)(((dim1 >> 16) & 0xFFFFu) | ((dim0 & 0xFFFFu) << 16)); // tile_dim0
  g1[4] = (int)(dim1 & 0xFFFFu);                    // tile_dim1 (tile_dim2 = 0)
  g1[5] = (int)stride0;                             // tensor_dim0_stride lo32
  g1[6] = 0;
  g1[7] = 0;
  i32x4 g2 = { 1, 1, 0, 0 };                        // tensor_dim2=1, dim3=1
  i32x4 g3 = { 0, 1 << 16, 0, 0 };                  // tensor_dim4=1
  i32x8 gx = { 0, 0, 0, 0, 0, 0, 0, 0 };            // unused trailing group
  if (store) __builtin_amdgcn_tensor_store_from_lds(g0, g1, g2, g3, gx, 0);
  else       __builtin_amdgcn_tensor_load_to_lds(g0, g1, g2, g3, gx, 0);
}

// ---------------------------------------------------------------------------
// Hoisted H16 A-operand signs: H16[m][k] = (-1)^popc(m&k), ±1.0f per lane.
//   A 16x4 layout: lanes 0-15 M=lane, VGPR g: K = g + 2*(lane>=16)
// ---------------------------------------------------------------------------
__device__ __forceinline__ void makeAsig(v2f* A) {
  const int lane = threadIdx.x & 31;
  const int n = lane & 15;
  const int kh = lane >> 4;
#pragma unroll
  for (int kc = 0; kc < 4; ++kc) {
    const int k0 = 4 * kc + 2 * kh;
    A[kc].x = (__popc(n & (k0 + 0)) & 1) ? -1.0f : 1.0f;
    A[kc].y = (__popc(n & (k0 + 1)) & 1) ? -1.0f : 1.0f;
  }
}

// One wave: V <- H16 * V for a 16x16 f32 tile in LDS, element (m,n) at
// lds[m*sM + n*sN] (float units). 4 chained V_WMMA_F32_16X16X4_F32.
__device__ __forceinline__ void hmul16(float* p, int sM, int sN, const v2f* As) {
  const int lane = threadIdx.x & 31;
  const int n = lane & 15;
  const int kh = lane >> 4;
  v2f B[4];
#pragma unroll
  for (int kc = 0; kc < 4; ++kc) {
    const int k0 = 4 * kc + 2 * kh;
    B[kc].x = p[(k0 + 0) * sM + n * sN];
    B[kc].y = p[(k0 + 1) * sM + n * sN];
  }
  v8f acc = {};
#pragma unroll
  for (int kc = 0; kc < 4; ++kc)
    acc = __builtin_amdgcn_wmma_f32_16x16x4_f32(false, As[kc], false, B[kc],
                                                (short)0, acc, false, false);
#pragma unroll
  for (int g = 0; g < 8; ++g)
    p[(g + 8 * kh) * sM + n * sN] = acc[g];
}

// 1024-point FWHT on one float plane in LDS; element m at base[m*es].
// Digits m = q*256 + b1*16 + b0: H16 over b0, H16 over b1 (WMMA), H4 over q.
// One wave executes this whole function.
__device__ void fwht1024(float* base, int es, const v2f* As) {
#pragma unroll
  for (int j = 0; j < 4; ++j) hmul16(base + 256 * es * j, es, 16 * es, As);
#pragma unroll
  for (int j = 0; j < 4; ++j) hmul16(base + 256 * es * j, 16 * es, es, As);
  const int lane = threadIdx.x & 31;
#pragma unroll
  for (int it = 0; it < 8; ++it) {
    const int c = it * 32 + lane;
    float* p = base + c * es;
    const float v0 = p[0], v1 = p[256 * es], v2 = p[512 * es], v3 = p[768 * es];
    p[0]        = v0 + v1 + v2 + v3;
    p[256 * es] = v0 - v1 + v2 - v3;
    p[512 * es] = v0 + v1 - v2 - v3;
    p[768 * es] = v0 - v1 - v2 + v3;
  }
}

// ---------------------------------------------------------------------------
// Diagonal factor tables: 6 x 1024 complex in ws.
// ev_j(m) = Chi_j[m>>10] * Clo_j[m&1023]; j0 carries i^popc, j2 carries
// (-i)^popc and the global 2^-20 scale.
// ---------------------------------------------------------------------------
__global__ void ktab(const float* __restrict__ ang, float2* __restrict__ tabs) {
  const int j = blockIdx.x;       // 0..5
  const int jj = j >> 1;          // diagonal 0,1,2
  const int part = j & 1;         // 0 = hi bits (qubits 0-9), 1 = lo
  for (int idx = threadIdx.x; idx < 1024; idx += blockDim.x) {
    float th = 0.0f;
#pragma unroll
    for (int q = 0; q < 10; ++q)
      th += ang[jj * 20 + (part ? (10 + q) : q)] * (float)((idx >> (9 - q)) & 1);
    float sn, cs;
    sincosf(0.5f * th, &sn, &cs);
    float re = cs, im = -sn;      // exp(-i*th/2)
    int p = 0;
    if (jj == 0) p = __popc(idx) & 3;
    else if (jj == 2) p = (4 - (__popc(idx) & 3)) & 3;
    float rr = re, ii = im;
    if (p == 1)      { rr = -im; ii =  re; }
    else if (p == 2) { rr = -re; ii = -im; }
    else if (p == 3) { rr =  im; ii = -re; }
    if (jj == 2 && part == 0) { rr *= 9.5367431640625e-07f; ii *= 9.5367431640625e-07f; }
    tabs[j * 1024 + idx] = make_float2(rr, ii);
  }
}

// ---------------------------------------------------------------------------
// K1: z2 = F_lo( D0[perm(v)] * x[perm(v)] ) -> out, natural order.
// Gather stays on VMEM (Gray-code pattern; hi paired with hi+512 so every
// cacheline is fully consumed inside the block). Output via two TDM stores.
// ---------------------------------------------------------------------------
__global__ __launch_bounds__(512) void k1(const float* __restrict__ xr,
                                          const float* __restrict__ xi,
                                          const float2* __restrict__ tabs,
                                          float2* __restrict__ out) {
  extern __shared__ float lds[];
  const unsigned h0 = blockIdx.x * 8;
  const size_t base = (size_t)blockIdx.y << NQ;
  const float2* Chi = tabs;
  const float2* Clo = tabs + 1024;
  v2f As[4];
  makeAsig(As);
  for (int e = threadIdx.x; e < 16384; e += 512) {
    const int seg = e >> 10, lo = e & 1023;
    const unsigned h = (seg < 8) ? (h0 + seg) : (h0 + 504 + seg);
    const unsigned m = permL((h << 10) | (unsigned)lo);
    const float2 a = Chi[m >> 10], bb = Clo[m & 1023];
    const c32 d = cmul({a.x, a.y}, {bb.x, bb.y});
    const c32 z = cmul({xr[base + m], xi[base + m]}, d);
    lds[2 * e] = z.x; lds[2 * e + 1] = z.y;
  }
  __syncthreads();
  const int w = threadIdx.x >> 5;
  fwht1024(lds + 2048 * w, 2, As);
  fwht1024(lds + 2048 * w + 1, 2, As);
  __syncthreads();
  if (threadIdx.x < 32) {
    const unsigned lbase = (unsigned)(size_t)lds;
    tdm_move(true, lbase,         out + base + ((size_t)h0 << 10),          8192, 1, 8192);
    tdm_move(true, lbase + 65536, out + base + ((size_t)(h0 + 512) << 10),  8192, 1, 8192);
    __builtin_amdgcn_s_wait_tensorcnt(0);
  }
}

// ---------------------------------------------------------------------------
// K2: z5 = F_hi( D1[fT(u)] * F_hi(z2) ), in place. Tile = 1024 hi x 16 lo,
// moved by TDM as a 2D tile (rows of 128B, stride 8KB). LDS keeps the natural
// [h][r] tile order; transforms run with element stride 32 floats.
// ---------------------------------------------------------------------------
__global__ __launch_bounds__(512) void k2(float2* __restrict__ buf,
                                          const float2* __restrict__ tabs) {
  extern __shared__ float lds[];
  const unsigned l0 = blockIdx.x * 16;
  const size_t base = (size_t)blockIdx.y << NQ;
  const float2* Chi = tabs + 2 * 1024;
  const float2* Clo = tabs + 3 * 1024;
  v2f As[4];
  makeAsig(As);
  if (threadIdx.x < 32) {
    tdm_move(false, (unsigned)(size_t)lds, buf + base + l0, 16, 1024, 1024);
    __builtin_amdgcn_s_wait_tensorcnt(0);
  }
  __syncthreads();
  const int w = threadIdx.x >> 5;                 // wave w owns lo-row r = w
  fwht1024(lds + 2 * w, 32, As);
  fwht1024(lds + 2 * w + 1, 32, As);
  __syncthreads();
  for (int e = threadIdx.x; e < 16384; e += 512) {
    const int h = e >> 4, r = e & 15;
    const unsigned m = fTmap(((unsigned)h << 10) | (l0 + r));
    const float2 a = Chi[m >> 10], bb = Clo[m & 1023];
    const c32 d = cmul({a.x, a.y}, {bb.x, bb.y});
    const int off = h * 32 + 2 * r;
    const c32 z = cmul({lds[off], lds[off + 1]}, d);
    lds[off] = z.x; lds[off + 1] = z.y;
  }
  __syncthreads();
  fwht1024(lds + 2 * w, 32, As);
  fwht1024(lds + 2 * w + 1, 32, As);
  __syncthreads();
  if (threadIdx.x < 32) {
    tdm_move(true, (unsigned)(size_t)lds, buf + base + l0, 16, 1024, 1024);
    __builtin_amdgcn_s_wait_tensorcnt(0);
  }
}

// ---------------------------------------------------------------------------
// K3: out[y] = D2[perm(y)] * 2^-20 * F_lo(z5)[y], in place.
// Tile = 16 hi x 1024 lo = one contiguous 128KB run: 1D TDM load/store.
// ---------------------------------------------------------------------------
__global__ __launch_bounds__(512) void k3(float2* __restrict__ buf,
                                          const float2* __restrict__ tabs) {
  extern __shared__ float lds[];
  const unsigned h0 = blockIdx.x * 16;
  const size_t base = (size_t)blockIdx.y << NQ;
  const float2* Chi = tabs + 4 * 1024;
  const float2* Clo = tabs + 5 * 1024;
  v2f As[4];
  makeAsig(As);
  if (threadIdx.x < 32) {
    tdm_move(false, (unsigned)(size_t)lds, buf + base + ((size_t)h0 << 10), 16384, 1, 16384);
    __builtin_amdgcn_s_wait_tensorcnt(0);
  }
  __syncthreads();
  const int w = threadIdx.x >> 5;
  fwht1024(lds + 2048 * w, 2, As);
  fwht1024(lds + 2048 * w + 1, 2, As);
  __syncthreads();
  for (int e = threadIdx.x; e < 16384; e += 512) {
    const int seg = e >> 10, lo = e & 1023;
    const unsigned y = ((h0 + (unsigned)seg) << 10) | (unsigned)lo;
    const unsigned m = permL(y);
    const float2 a = Chi[m >> 10], bb = Clo[m & 1023];
    const c32 d = cmul({a.x, a.y}, {bb.x, bb.y});
    const c32 z = cmul({lds[2 * e], lds[2 * e + 1]}, d);
    lds[2 * e] = z.x; lds[2 * e + 1] = z.y;
  }
  __syncthreads();
  if (threadIdx.x < 32) {
    tdm_move(true, (unsigned)(size_t)lds, buf + base + ((size_t)h0 << 10), 16384, 1, 16384);
    __builtin_amdgcn_s_wait_tensorcnt(0);
  }
}

extern "C" void kernel_launch(void* const* d_in, const int* in_sizes, int n_in,
                              void* d_out, int out_size, void* d_ws, size_t ws_size,
                              hipStream_t stream) {
  const float* xr  = (const float*)d_in[0];
  const float* xi  = (const float*)d_in[1];
  const float* ang = (const float*)d_in[2];
  float2* tabs = (float2*)d_ws;            // 6 * 1024 * 8 B = 48 KB
  float2* out  = (float2*)d_out;           // interleaved complex64, also scratch

  ktab<<<6, 256, 0, stream>>>(ang, tabs);

  const dim3 grid(64, BATCH);
  const size_t ldsb = 16 * 1024 * sizeof(float2);   // 128 KB dynamic LDS
  k1<<<grid, 512, ldsb, stream>>>(xr, xi, tabs, out);
  k2<<<grid, 512, ldsb, stream>>>(out, tabs);
  k3<<<grid, 512, ldsb, stream>>>(out, tabs);
}